// GAT_22634477650523
// MI455X (gfx1250) — compile-verified
//
#include <hip/hip_runtime.h>

#define HH   8     // heads (conv1)
#define CC   16    // hidden channels
#define HCC  128   // HH*CC
#define CO   32    // conv2 out channels

typedef __attribute__((ext_vector_type(2))) float    v2f;
typedef __attribute__((ext_vector_type(8))) float    v8f;
typedef __attribute__((ext_vector_type(4))) unsigned u32x4;
typedef __attribute__((ext_vector_type(8))) int      i32x8;
typedef __attribute__((ext_vector_type(4))) int      i32x4;

__device__ __forceinline__ unsigned fenc(float f) {
    unsigned u = __float_as_uint(f);
    return (u & 0x80000000u) ? ~u : (u | 0x80000000u);
}
__device__ __forceinline__ float fdec(unsigned e) {
    return __uint_as_float((e & 0x80000000u) ? (e & 0x7fffffffu) : ~e);
}
__device__ __forceinline__ float lrelu(float x) { return x > 0.f ? x : 0.2f * x; }

// ---- K1: mean/var stats of x over N (wave-reduce + atomics) ----
__global__ void k_bn0_stats(const float* __restrict__ x, int n, float* __restrict__ small) {
    int i = blockIdx.x * blockDim.x + threadIdx.x;
    float v = (i < n) ? x[i] : 0.f;
    float s = v, q = v * v;
#pragma unroll
    for (int m = 1; m < 32; m <<= 1) { s += __shfl_xor(s, m, 32); q += __shfl_xor(q, m, 32); }
    if ((threadIdx.x & 31) == 0) { atomicAdd(&small[0], s); atomicAdd(&small[1], q); }
}

// ---- K2: finalize bn0 affine; precompute per-head attn scalars ws1/wd1 ----
__global__ void k_prep1(const float* __restrict__ W1, const float* __restrict__ a_src1,
                        const float* __restrict__ a_dst1, const float* __restrict__ g0,
                        const float* __restrict__ b0, int n, float* __restrict__ small) {
    int t = threadIdx.x;
    if (t == 0) {
        float m    = small[0] / (float)n;
        float var  = small[1] / (float)n - m * m;
        float rstd = rsqrtf(var + 1e-5f);
        float sc   = rstd * g0[0];
        small[2] = sc;
        small[3] = b0[0] - m * sc;
    }
    if (t < HH) {
        float ws = 0.f, wd = 0.f;
        for (int c = 0; c < CC; ++c) {
            float w = W1[t * CC + c];
            ws += w * a_src1[t * CC + c];
            wd += w * a_dst1[t * CC + c];
        }
        small[8 + t]  = ws;
        small[16 + t] = wd;
    }
}

// ---- K3: normalized x, per-node per-head attn terms ----
__global__ void k_node1(const float* __restrict__ x, int n, const float* __restrict__ small,
                        float* __restrict__ xh, float* __restrict__ as1, float* __restrict__ ad1) {
    int i = blockIdx.x * blockDim.x + threadIdx.x;
    if (i >= n) return;
    float v = x[i] * small[2] + small[3];
    xh[i] = v;
#pragma unroll
    for (int h = 0; h < HH; ++h) {
        as1[i * HH + h] = v * small[8 + h];
        ad1[i * HH + h] = v * small[16 + h];
    }
}

// ---- K4: conv1 segment-max over dst (thread per edge*head) ----
__global__ void k_edge1_max(const int* __restrict__ ei, int E, int n,
                            const float* __restrict__ as1, const float* __restrict__ ad1,
                            unsigned* __restrict__ m1) {
    int t = blockIdx.x * blockDim.x + threadIdx.x;
    int tot = (E + n) * HH;
    if (t >= tot) return;
    int e = t >> 3, h = t & 7;
    int s, d;
    if (e < E) { s = ei[e]; d = ei[E + e]; } else { s = e - E; d = s; }
    float sc = lrelu(as1[s * HH + h] + ad1[d * HH + h]);
    atomicMax(&m1[d * HH + h], fenc(sc));
}

// ---- K5: conv1 exp-sum + weighted numerator (scalar messages!) ----
__global__ void k_edge1_sum(const int* __restrict__ ei, int E, int n,
                            const float* __restrict__ as1, const float* __restrict__ ad1,
                            const unsigned* __restrict__ m1, const float* __restrict__ xh,
                            float* __restrict__ sum1, float* __restrict__ num1) {
    int t = blockIdx.x * blockDim.x + threadIdx.x;
    int tot = (E + n) * HH;
    if (t >= tot) return;
    int e = t >> 3, h = t & 7;
    int s, d;
    if (e < E) { s = ei[e]; d = ei[E + e]; } else { s = e - E; d = s; }
    int di = d * HH + h;
    float sc = lrelu(as1[s * HH + h] + ad1[di]);
    float ex = expf(sc - fdec(m1[di]));
    atomicAdd(&sum1[di], ex);
    atomicAdd(&num1[di], ex * xh[s]);
}

// ---- K6: s1 = num/den; accumulate per-head mean/var stats ----
__global__ void k_node1b(int n, const float* __restrict__ sum1, const float* __restrict__ num1,
                         float* __restrict__ s1, float* __restrict__ small) {
    int t = blockIdx.x * blockDim.x + threadIdx.x;
    int tot = n * HH;
    float s = 0.f;
    if (t < tot) { s = num1[t] / (sum1[t] + 1e-16f); s1[t] = s; }
    float rs = s, rq = s * s;
#pragma unroll
    for (int m = 8; m < 32; m <<= 1) { rs += __shfl_xor(rs, m, 32); rq += __shfl_xor(rq, m, 32); }
    int lane = threadIdx.x & 31;
    if (lane < 8) { atomicAdd(&small[24 + lane], rs); atomicAdd(&small[32 + lane], rq); }
}

// ---- K7: fold bn1 into affine coefficients A[hc], B[hc] ----
__global__ void k_prep2(const float* __restrict__ W1, const float* __restrict__ g1,
                        const float* __restrict__ bb1, int n, float* __restrict__ small) {
    int hc = threadIdx.x;
    if (hc >= HCC) return;
    int h = hc >> 4;
    float mean = small[24 + h] / (float)n;
    float var  = small[32 + h] / (float)n - mean * mean;
    float w    = W1[hc];
    float inv  = rsqrtf(var * w * w + 1e-5f);
    small[64 + hc]  = w * g1[hc] * inv;                    // A
    small[192 + hc] = -mean * w * g1[hc] * inv + bb1[hc];  // B
}

// ---- K8: materialize h2in = relu(A*s1 + B)  [N,128] ----
__global__ void k_node2(int n, const float* __restrict__ s1, const float* __restrict__ small,
                        float* __restrict__ h2in) {
    long long t = (long long)blockIdx.x * blockDim.x + threadIdx.x;
    if (t >= (long long)n * HCC) return;
    int hc = (int)(t & (HCC - 1));
    int nn = (int)(t >> 7);
    int h  = hc >> 4;
    float v = small[64 + hc] * s1[nn * HH + h] + small[192 + hc];
    h2in[t] = fmaxf(v, 0.f);
}

// ---- K9: h2[N,32] = h2in[N,128] @ W2[128,32]  via V_WMMA_F32_16X16X4_F32 ----
// W2 (16KB) staged into LDS once per workgroup via the Tensor Data Mover;
// one wave per 16-row tile; 2 column tiles (32 cols); K=128 in steps of 4.
__global__ void k_gemm_wmma(int n, const float* __restrict__ h2in,
                            const float* __restrict__ W2, float* __restrict__ h2) {
    __shared__ float w2s[HCC * CO];   // 16 KB

#if __has_builtin(__builtin_amdgcn_tensor_load_to_lds)
    if (threadIdx.x < 32) {
        // 1-D TDM copy: 4096 f32 elements, contiguous. LDS byte address is the
        // low 32 bits of the generic pointer.
        unsigned lds_off = (unsigned)(unsigned long long)(const void*)w2s;
        unsigned long long ga = (unsigned long long)W2;
        u32x4 g0;
        g0.x = 1u;                                   // count=1, user descriptor
        g0.y = lds_off;                              // lds_addr
        g0.z = (unsigned)ga;                         // global_addr[31:0]
        g0.w = (unsigned)(ga >> 32) | 0x80000000u;   // global_addr[56:32] | type=2
        i32x8 g1;
        g1[0] = 0x20000;                             // data_size = 4 bytes
        g1[1] = (int)(4096u << 16);                  // tensor_dim0 = 4096
        g1[2] = (int)(1u << 16);                     // tensor_dim1 = 1
        g1[3] = (int)(4096u << 16);                  // tile_dim0 = 4096
        g1[4] = 1;                                   // tile_dim1 = 1
        g1[5] = 4096;                                // tensor_dim0_stride = 4096
        g1[6] = 0;
        g1[7] = 0;
        i32x4 g2 = {0, 0, 0, 0};
        i32x4 g3 = {0, 0, 0, 0};
#if __clang_major__ >= 23
        i32x8 g4 = {0, 0, 0, 0, 0, 0, 0, 0};
        __builtin_amdgcn_tensor_load_to_lds(g0, g1, g2, g3, g4, 0);
#else
        __builtin_amdgcn_tensor_load_to_lds(g0, g1, g2, g3, 0);
#endif
        __builtin_amdgcn_s_wait_tensorcnt(0);
    }
#else
    for (int i = threadIdx.x; i < HCC * CO; i += blockDim.x) w2s[i] = W2[i];
#endif
    __syncthreads();

    int wave = (blockIdx.x * blockDim.x + threadIdx.x) >> 5;
    int lane = threadIdx.x & 31;
    int m0 = wave * 16;
    if (m0 >= n) return;
    int lh = lane >> 4;   // 0: K pair {0,1}   1: K pair {2,3}
    int lm = lane & 15;
    int r  = m0 + lm; if (r >= n) r = n - 1;   // clamp loads, guard stores
    const float* arow = h2in + (long long)r * HCC;
    v8f acc0 = {};
    v8f acc1 = {};
    for (int k0 = 0; k0 < HCC; k0 += 4) {
        int ka = k0 + 2 * lh;
        v2f a;  a.x  = arow[ka];                  a.y  = arow[ka + 1];
        v2f b0; b0.x = w2s[ka * CO + lm];         b0.y = w2s[(ka + 1) * CO + lm];
        v2f b1; b1.x = w2s[ka * CO + 16 + lm];    b1.y = w2s[(ka + 1) * CO + 16 + lm];
        acc0 = __builtin_amdgcn_wmma_f32_16x16x4_f32(false, a, false, b0, (short)0, acc0, false, false);
        acc1 = __builtin_amdgcn_wmma_f32_16x16x4_f32(false, a, false, b1, (short)0, acc1, false, false);
    }
#pragma unroll
    for (int v = 0; v < 8; ++v) {
        int row = m0 + v + 8 * lh;
        if (row < n) {
            h2[row * CO + lm]      = acc0[v];
            h2[row * CO + 16 + lm] = acc1[v];
        }
    }
}

// ---- K10: conv2 attention terms (wave per node, lane = channel) ----
__global__ void k_attn2(int n, const float* __restrict__ h2, const float* __restrict__ a_src2,
                        const float* __restrict__ a_dst2, float* __restrict__ as2,
                        float* __restrict__ ad2) {
    long long g = (long long)blockIdx.x * blockDim.x + threadIdx.x;
    int node = (int)(g >> 5);
    int lane = threadIdx.x & 31;
    if (node >= n) return;
    float v  = h2[node * CO + lane];
    float ps = v * a_src2[lane];
    float pd = v * a_dst2[lane];
#pragma unroll
    for (int m = 1; m < 32; m <<= 1) { ps += __shfl_xor(ps, m, 32); pd += __shfl_xor(pd, m, 32); }
    if (lane == 0) { as2[node] = ps; ad2[node] = pd; }
}

// ---- K11: conv2 segment-max (thread per edge) ----
__global__ void k_edge2_max(const int* __restrict__ ei, int E, int n,
                            const float* __restrict__ as2, const float* __restrict__ ad2,
                            unsigned* __restrict__ m2) {
    int e = blockIdx.x * blockDim.x + threadIdx.x;
    if (e >= E + n) return;
    int s, d;
    if (e < E) { s = ei[e]; d = ei[E + e]; } else { s = e - E; d = s; }
    atomicMax(&m2[d], fenc(lrelu(as2[s] + ad2[d])));
}

// ---- K12: conv2 exp-sum + 32-wide weighted scatter (wave per edge) ----
__global__ void k_edge2_sum(const int* __restrict__ ei, int E, int n,
                            const float* __restrict__ as2, const float* __restrict__ ad2,
                            const unsigned* __restrict__ m2, const float* __restrict__ h2,
                            float* __restrict__ sum2, float* __restrict__ num2) {
    long long g = (long long)blockIdx.x * blockDim.x + threadIdx.x;
    int e = (int)(g >> 5);
    int lane = threadIdx.x & 31;
    if (e >= E + n) return;
    int s, d;
    if (e < E) { s = ei[e]; d = ei[E + e]; } else { s = e - E; d = s; }
    float ex = expf(lrelu(as2[s] + ad2[d]) - fdec(m2[d]));
    if (lane == 0) atomicAdd(&sum2[d], ex);
    atomicAdd(&num2[d * CO + lane], ex * h2[s * CO + lane]);
}

// ---- K13: normalize + bias + relu + MLP head (thread per node) ----
__global__ void k_mlp(int n, const float* __restrict__ num2, const float* __restrict__ sum2,
                      const float* __restrict__ b2, const float* __restrict__ lw1,
                      const float* __restrict__ lb1, const float* __restrict__ lw2,
                      const float* __restrict__ lb2, float* __restrict__ out) {
    int i = blockIdx.x * blockDim.x + threadIdx.x;
    if (i >= n) return;
    float o[CO];
    float inv = 1.f / (sum2[i] + 1e-16f);
#pragma unroll
    for (int c = 0; c < CO; ++c) o[c] = fmaxf(num2[i * CO + c] * inv + b2[c], 0.f);
    float y = lb2[0];
#pragma unroll
    for (int k = 0; k < CC; ++k) {
        float t = lb1[k];
#pragma unroll
        for (int c = 0; c < CO; ++c) t += o[c] * lw1[c * CC + k];
        y += fmaxf(t, 0.f) * lw2[k];
    }
    out[i] = y;
}

extern "C" void kernel_launch(void* const* d_in, const int* in_sizes, int n_in,
                              void* d_out, int out_size, void* d_ws, size_t ws_size,
                              hipStream_t stream) {
    const float* x      = (const float*)d_in[0];
    const int*   ei     = (const int*)d_in[1];
    const float* W1     = (const float*)d_in[3];
    const float* a_src1 = (const float*)d_in[4];
    const float* a_dst1 = (const float*)d_in[5];
    const float* W2     = (const float*)d_in[7];
    const float* a_src2 = (const float*)d_in[8];
    const float* a_dst2 = (const float*)d_in[9];
    const float* b2     = (const float*)d_in[10];
    const float* bn0_g  = (const float*)d_in[11];
    const float* bn0_b  = (const float*)d_in[12];
    const float* bn1_g  = (const float*)d_in[13];
    const float* bn1_b  = (const float*)d_in[14];
    const float* lw1    = (const float*)d_in[15];
    const float* lb1    = (const float*)d_in[16];
    const float* lw2    = (const float*)d_in[17];
    const float* lb2    = (const float*)d_in[18];
    float* out = (float*)d_out;

    int n = in_sizes[0];
    int E = in_sizes[1] / 2;
    int E2 = E + n;

    float* ws = (float*)d_ws;
    size_t off = 512;
    float*    small = ws;
    float*    xh   = ws + off; off += (size_t)n;
    float*    as1  = ws + off; off += (size_t)8 * n;
    float*    ad1  = ws + off; off += (size_t)8 * n;
    unsigned* m1   = (unsigned*)(ws + off); off += (size_t)8 * n;
    float*    sum1 = ws + off; off += (size_t)8 * n;
    float*    num1 = ws + off; off += (size_t)8 * n;
    float*    s1   = ws + off; off += (size_t)8 * n;
    float*    h2in = ws + off; off += (size_t)128 * n;
    float*    h2   = ws + off; off += (size_t)32 * n;
    float*    as2  = ws + off; off += (size_t)n;
    float*    ad2  = ws + off; off += (size_t)n;
    unsigned* m2   = (unsigned*)(ws + off); off += (size_t)n;
    float*    sum2 = ws + off; off += (size_t)n;
    float*    num2 = ws + off; off += (size_t)32 * n;

    hipMemsetAsync(d_ws, 0, off * sizeof(float), stream);

    const int B = 256;
    k_bn0_stats<<<(n + B - 1) / B, B, 0, stream>>>(x, n, small);
    k_prep1<<<1, 64, 0, stream>>>(W1, a_src1, a_dst1, bn0_g, bn0_b, n, small);
    k_node1<<<(n + B - 1) / B, B, 0, stream>>>(x, n, small, xh, as1, ad1);

    long long t1 = (long long)E2 * HH;
    k_edge1_max<<<(unsigned)((t1 + B - 1) / B), B, 0, stream>>>(ei, E, n, as1, ad1, m1);
    k_edge1_sum<<<(unsigned)((t1 + B - 1) / B), B, 0, stream>>>(ei, E, n, as1, ad1, m1, xh, sum1, num1);

    k_node1b<<<((n * HH) + B - 1) / B, B, 0, stream>>>(n, sum1, num1, s1, small);
    k_prep2<<<1, 128, 0, stream>>>(W1, bn1_g, bn1_b, n, small);

    long long t2 = (long long)n * HCC;
    k_node2<<<(unsigned)((t2 + B - 1) / B), B, 0, stream>>>(n, s1, small, h2in);

    int tiles = (n + 15) / 16;
    k_gemm_wmma<<<(tiles + 7) / 8, B, 0, stream>>>(n, h2in, W2, h2);

    long long t3 = (long long)n * 32;
    k_attn2<<<(unsigned)((t3 + B - 1) / B), B, 0, stream>>>(n, h2, a_src2, a_dst2, as2, ad2);

    k_edge2_max<<<(E2 + B - 1) / B, B, 0, stream>>>(ei, E, n, as2, ad2, m2);
    long long t4 = (long long)E2 * 32;
    k_edge2_sum<<<(unsigned)((t4 + B - 1) / B), B, 0, stream>>>(ei, E, n, as2, ad2, m2, h2, sum2, num2);

    k_mlp<<<(n + B - 1) / B, B, 0, stream>>>(n, num2, sum2, b2, lw1, lb1, lw2, lb2, out);
}